// NodeEmbedding_41669772706306
// MI455X (gfx1250) — compile-verified
//
#include <hip/hip_runtime.h>

typedef __attribute__((ext_vector_type(2))) float v2f;
typedef __attribute__((ext_vector_type(8))) float v8f;

#define DOUT 64

// ---------------------------------------------------------------------------
// deg[i] = 1.0 (self loop)
__global__ void k_deg_init(float* __restrict__ deg, int n) {
    int i = blockIdx.x * blockDim.x + threadIdx.x;
    if (i < n) deg[i] = 1.0f;
}

// deg[col[e]] += w[e]
__global__ void k_deg_accum(const long long* __restrict__ ei,
                            const float* __restrict__ ew,
                            float* __restrict__ deg, int E) {
    int e = blockIdx.x * blockDim.x + threadIdx.x;
    if (e < E) atomicAdd(deg + (int)ei[(size_t)E + e], ew[e]);
}

// in place: deg -> dinv
__global__ void k_dinv(float* __restrict__ deg, int n) {
    int i = blockIdx.x * blockDim.x + threadIdx.x;
    if (i < n) {
        float d = deg[i];
        deg[i] = (d > 0.0f) ? rsqrtf(d) : 0.0f;
    }
}

// per-edge norm + compacted int32 indices
__global__ void k_norm(const long long* __restrict__ ei,
                       const float* __restrict__ ew,
                       const float* __restrict__ dinv,
                       float* __restrict__ norm,
                       int2* __restrict__ rowcol, int E) {
    int e = blockIdx.x * blockDim.x + threadIdx.x;
    if (e < E) {
        int r = (int)ei[e];
        int c = (int)ei[(size_t)E + e];
        norm[e] = dinv[r] * ew[e] * dinv[c];
        rowcol[e] = make_int2(r, c);
    }
}

// ---------------------------------------------------------------------------
// h = x @ W via V_WMMA_F32_16X16X4_F32.  One wave -> 16 rows x 64 cols.
// A 16x4 f32 layout: lane l, vgpr v  -> M = l%16, K = k0 + v + 2*(l/16)
// B 4x16 f32 layout: lane l, vgpr v  -> K = k0 + v + 2*(l/16), N = l%16
// C/D 16x16 f32:     lane l, vgpr r  -> M = r + 8*(l/16),      N = l%16
__global__ void k_gemm_wmma(const float* __restrict__ x,
                            const float* __restrict__ W,
                            float* __restrict__ h, int N) {
    int wave = (blockIdx.x * blockDim.x + threadIdx.x) >> 5;
    int lane = threadIdx.x & 31;
    int m0 = wave * 16;
    if (m0 >= N) return;            // wave-uniform: whole wave exits, EXEC stays full

    const int half = lane >> 4;     // 0 or 1
    const int l16  = lane & 15;

    v8f acc0 = {}, acc1 = {}, acc2 = {}, acc3 = {};
    const float* xrow = x + (size_t)(m0 + l16) * DOUT;

#pragma unroll
    for (int k0 = 0; k0 < DOUT; k0 += 4) {
        int ka = k0 + 2 * half;
        v2f a;
        a.x = xrow[ka + 0];
        a.y = xrow[ka + 1];

        v2f b0, b1, b2, b3;
        const float* w0 = W + (size_t)(ka + 0) * DOUT + l16;
        const float* w1 = W + (size_t)(ka + 1) * DOUT + l16;
        b0.x = w0[0];  b0.y = w1[0];
        b1.x = w0[16]; b1.y = w1[16];
        b2.x = w0[32]; b2.y = w1[32];
        b3.x = w0[48]; b3.y = w1[48];

        acc0 = __builtin_amdgcn_wmma_f32_16x16x4_f32(false, a, false, b0, (short)0, acc0, false, false);
        acc1 = __builtin_amdgcn_wmma_f32_16x16x4_f32(false, a, false, b1, (short)0, acc1, false, false);
        acc2 = __builtin_amdgcn_wmma_f32_16x16x4_f32(false, a, false, b2, (short)0, acc2, false, false);
        acc3 = __builtin_amdgcn_wmma_f32_16x16x4_f32(false, a, false, b3, (short)0, acc3, false, false);
    }

#pragma unroll
    for (int r = 0; r < 8; ++r) {
        size_t row = (size_t)(m0 + r + 8 * half) * DOUT + l16;
        h[row +  0] = acc0[r];
        h[row + 16] = acc1[r];
        h[row + 32] = acc2[r];
        h[row + 48] = acc3[r];
    }
}

// ---------------------------------------------------------------------------
// out[i][d] = h[i][d] * dinv[i]^2 + b[d]   (self-loop + bias; also inits out)
__global__ void k_out_init(const float* __restrict__ h,
                           const float* __restrict__ dinv,
                           const float* __restrict__ b,
                           float* __restrict__ out, int total) {
    int t = blockIdx.x * blockDim.x + threadIdx.x;
    if (t < total) {
        int i = t >> 6;
        int d = t & 63;
        float s = dinv[i];
        out[t] = h[t] * s * s + b[d];
    }
}

// out[col][d] += h[row][d] * norm[e]   (64 lanes per edge, f32 L2 atomics)
__global__ void k_scatter(const float* __restrict__ h,
                          const float* __restrict__ norm,
                          const int2* __restrict__ rowcol,
                          float* __restrict__ out, int total) {
    int t = blockIdx.x * blockDim.x + threadIdx.x;
    if (t < total) {
        int e = t >> 6;
        int d = t & 63;
        int2 rc = rowcol[e];
        atomicAdd(out + (size_t)rc.y * DOUT + d,
                  h[(size_t)rc.x * DOUT + d] * norm[e]);
    }
}

__global__ void k_relu(float* __restrict__ out, int n) {
    int i = blockIdx.x * blockDim.x + threadIdx.x;
    if (i < n) out[i] = fmaxf(out[i], 0.0f);
}

// ---------------------------------------------------------------------------
extern "C" void kernel_launch(void* const* d_in, const int* in_sizes, int n_in,
                              void* d_out, int out_size, void* d_ws, size_t ws_size,
                              hipStream_t stream) {
    const float*     x  = (const float*)d_in[0];     // [N, 64] f32
    const long long* ei = (const long long*)d_in[1]; // [2, E] i64
    const float*     ew = (const float*)d_in[2];     // [E] f32
    const float*     W  = (const float*)d_in[3];     // [64, 64] f32
    const float*     b  = (const float*)d_in[4];     // [64] f32
    float* out = (float*)d_out;

    const int N = in_sizes[0] / DOUT;   // 50000
    const int E = in_sizes[2];          // 800000
    const int ND = N * DOUT;

    // workspace carve-up (16B aligned)
    auto align16 = [](size_t v) { return (v + 15) & ~(size_t)15; };
    char* ws = (char*)d_ws;
    float* h    = (float*)ws;                       size_t off = align16((size_t)ND * 4);
    float* dinv = (float*)(ws + off);               off += align16((size_t)N * 4);
    float* norm = (float*)(ws + off);               off += align16((size_t)E * 4);
    int2*  rc   = (int2*)(ws + off);

    const int B = 256;

    k_deg_init <<<(N + B - 1) / B, B, 0, stream>>>(dinv, N);
    k_deg_accum<<<(E + B - 1) / B, B, 0, stream>>>(ei, ew, dinv, E);
    k_dinv     <<<(N + B - 1) / B, B, 0, stream>>>(dinv, N);
    k_norm     <<<(E + B - 1) / B, B, 0, stream>>>(ei, ew, dinv, norm, rc, E);

    // one wave per 16-row tile; 8 waves per 256-thread block
    int waves = (N + 15) / 16;
    int gblocks = (waves + 7) / 8;
    k_gemm_wmma<<<gblocks, B, 0, stream>>>(x, W, h, N);

    k_out_init <<<(ND + B - 1) / B, B, 0, stream>>>(h, dinv, b, out, ND);

    int total = E * DOUT;   // 51.2M lane-tasks
    k_scatter  <<<(total + B - 1) / B, B, 0, stream>>>(h, norm, rc, out, total);

    k_relu     <<<(ND + B - 1) / B, B, 0, stream>>>(out, ND);
}